// Custom_Loss_84937273246180
// MI455X (gfx1250) — compile-verified
//
#include <hip/hip_runtime.h>
#include <cstdint>
#include <cstddef>

// Problem size (fixed by the reference: 8192 x 8192 f32 matrices).
#define N 8192

constexpr int TPB          = 256;        // threads per block (8 waves, wave32)
constexpr int CPT          = 4;          // columns per thread (float4 / b128)
constexpr int COLS_PER_BLK = TPB * CPT;  // 1024 columns per block
constexpr int ROWS_PER_BLK = 128;        // row slice per block
constexpr int STAGES       = 8;          // async copies in flight per wave
constexpr int SLOTS        = 2 * STAGES; // LDS ring slots (avoids read/refill hazard)
static_assert(STAGES == 8, "s_wait_asynccnt immediates below assume STAGES == 8");

// Monotone key: f32 -> u32 such that key(a) < key(b) iff a < b (for non-NaN).
__device__ __forceinline__ unsigned fkey(float v) {
    unsigned b = __float_as_uint(v);
    return (b & 0x80000000u) ? ~b : (b | 0x80000000u);
}

// ---------------------------------------------------------------------------
// Kernel 1: zero the per-column packed (key, ~row) workspace (identity of max).
// ---------------------------------------------------------------------------
__global__ void init_ws(unsigned long long* __restrict__ packed) {
    int i = blockIdx.x * blockDim.x + threadIdx.x;
    if (i < N) packed[i] = 0ull;
}

// ---------------------------------------------------------------------------
// Kernel 2: column-wise partial argmax over a 1024-col x 128-row tile,
// streamed through LDS with CDNA5 async global->LDS copies (ASYNCcnt path),
// non-temporal (single-use 256MB stream; don't churn the 192MB L2).
// Each lane stages and consumes its own 16B, so no workgroup barriers needed.
// ---------------------------------------------------------------------------
__global__ __launch_bounds__(TPB) void argmax_cols(const float* __restrict__ A,
                                                   unsigned long long* __restrict__ packed) {
    __shared__ float smem[SLOTS * COLS_PER_BLK];   // 16 * 4KB = 64KB

    const int t  = threadIdx.x;
    const int c0 = blockIdx.x * COLS_PER_BLK + CPT * t;  // first of 4 cols
    const int r0 = blockIdx.y * ROWS_PER_BLK;            // first row of slice

    const float*   gbase    = A + (size_t)r0 * N + c0;
    // Low 32 bits of the generic address of LDS == wave-relative LDS byte offset.
    const uint32_t lds_base = (uint32_t)(uintptr_t)(&smem[0]) +
                              (uint32_t)(CPT * sizeof(float)) * (uint32_t)t;

    // Prologue: fill the pipeline with STAGES async b128 copies.
#pragma unroll
    for (int s = 0; s < STAGES; ++s) {
        uint32_t     ldsa = lds_base + (uint32_t)(s * COLS_PER_BLK * sizeof(float));
        const float* g    = gbase + (size_t)s * N;
        asm volatile("global_load_async_to_lds_b128 %0, %1, off th:TH_LOAD_NT"
                     :: "v"(ldsa), "v"(g) : "memory");
    }

    float m0 = -3.402823466e38f, m1 = m0, m2 = m0, m3 = m0;
    int   i0 = 0, i1 = 0, i2 = 0, i3 = 0;

    // Steady state: wait for the oldest copy, consume it, refill ring slot r+STAGES.
    for (int r = 0; r < ROWS_PER_BLK - STAGES; ++r) {
        asm volatile("s_wait_asynccnt 0x7" ::: "memory");  // oldest of 8 in flight done
        const float4 v = *reinterpret_cast<const float4*>(
            &smem[(r & (SLOTS - 1)) * COLS_PER_BLK + CPT * t]);

        {   // refill: row r+STAGES into ring slot (r+STAGES) % SLOTS
            const int    rn   = r + STAGES;
            uint32_t     ldsa = lds_base +
                (uint32_t)((rn & (SLOTS - 1)) * COLS_PER_BLK * sizeof(float));
            const float* g    = gbase + (size_t)rn * N;
            asm volatile("global_load_async_to_lds_b128 %0, %1, off th:TH_LOAD_NT"
                         :: "v"(ldsa), "v"(g) : "memory");
        }

        const int row = r0 + r;
        if (v.x > m0) { m0 = v.x; i0 = row; }   // strict '>' => first occurrence wins
        if (v.y > m1) { m1 = v.y; i1 = row; }
        if (v.z > m2) { m2 = v.z; i2 = row; }
        if (v.w > m3) { m3 = v.w; i3 = row; }
    }

    // Tail: drain the remaining STAGES copies.
    asm volatile("s_wait_asynccnt 0x0" ::: "memory");
#pragma unroll
    for (int r = ROWS_PER_BLK - STAGES; r < ROWS_PER_BLK; ++r) {
        const float4 v = *reinterpret_cast<const float4*>(
            &smem[(r & (SLOTS - 1)) * COLS_PER_BLK + CPT * t]);
        const int row = r0 + r;
        if (v.x > m0) { m0 = v.x; i0 = row; }
        if (v.y > m1) { m1 = v.y; i1 = row; }
        if (v.z > m2) { m2 = v.z; i2 = row; }
        if (v.w > m3) { m3 = v.w; i3 = row; }
    }

    // Merge partials: pack (key << 32) | (~row). max() picks the larger value,
    // and on equal values the smaller row index (first occurrence). u64 max is
    // associative/commutative/idempotent -> deterministic regardless of order.
    unsigned long long p;
    p = ((unsigned long long)fkey(m0) << 32) | (unsigned long long)(0xFFFFFFFFu - (unsigned)i0);
    atomicMax(&packed[c0 + 0], p);
    p = ((unsigned long long)fkey(m1) << 32) | (unsigned long long)(0xFFFFFFFFu - (unsigned)i1);
    atomicMax(&packed[c0 + 1], p);
    p = ((unsigned long long)fkey(m2) << 32) | (unsigned long long)(0xFFFFFFFFu - (unsigned)i2);
    atomicMax(&packed[c0 + 2], p);
    p = ((unsigned long long)fkey(m3) << 32) | (unsigned long long)(0xFFFFFFFFu - (unsigned)i3);
    atomicMax(&packed[c0 + 3], p);
}

// ---------------------------------------------------------------------------
// Kernel 3: decode path, gather cost[path[i], path[(i+1)%N]], per-block
// fixed-order tree reduction into partials (32 blocks, one gather per thread).
// ---------------------------------------------------------------------------
__global__ __launch_bounds__(256) void tour_partial(const float* __restrict__ C,
                                                    const unsigned long long* __restrict__ packed,
                                                    float* __restrict__ partials) {
    __shared__ float red[256];
    const int t = threadIdx.x;
    const int i = blockIdx.x * 256 + t;             // i in [0, N)

    const unsigned pi = 0xFFFFFFFFu - (unsigned)(packed[i] & 0xFFFFFFFFull);
    const unsigned pj = 0xFFFFFFFFu - (unsigned)(packed[(i + 1) & (N - 1)] & 0xFFFFFFFFull);
    red[t] = C[(size_t)pi * N + pj];
    __syncthreads();
    for (int s = 128; s > 0; s >>= 1) {
        if (t < s) red[t] += red[t + s];
        __syncthreads();
    }
    if (t == 0) partials[blockIdx.x] = red[0];
}

// ---------------------------------------------------------------------------
// Kernel 4: fixed serial order over 32 partials -> deterministic scalar out.
// ---------------------------------------------------------------------------
__global__ void tour_final(const float* __restrict__ partials, float* __restrict__ out) {
    if (threadIdx.x == 0) {
        float s = 0.0f;
        for (int b = 0; b < N / 256; ++b) s += partials[b];
        out[0] = s;
    }
}

// ---------------------------------------------------------------------------
// Host-side launch. d_in[0] = solution_matrix (f32 NxN), d_in[1] = cost_matrix
// (f32 NxN), d_out = 1 float. Workspace: N*8 B packed argmax + 32 f32 partials.
// ---------------------------------------------------------------------------
extern "C" void kernel_launch(void* const* d_in, const int* in_sizes, int n_in,
                              void* d_out, int out_size, void* d_ws, size_t ws_size,
                              hipStream_t stream) {
    (void)in_sizes; (void)n_in; (void)out_size; (void)ws_size;

    const float* A   = (const float*)d_in[0];
    const float* C   = (const float*)d_in[1];
    float*       out = (float*)d_out;
    unsigned long long* packed   = (unsigned long long*)d_ws;   // N * 8 bytes
    float*              partials = (float*)(packed + N);        // 32 floats

    init_ws<<<dim3(N / 256), dim3(256), 0, stream>>>(packed);

    dim3 grid(N / COLS_PER_BLK, N / ROWS_PER_BLK);  // 8 x 64 = 512 blocks
    argmax_cols<<<grid, dim3(TPB), 0, stream>>>(A, packed);

    tour_partial<<<dim3(N / 256), dim3(256), 0, stream>>>(C, packed, partials);
    tour_final<<<dim3(1), dim3(32), 0, stream>>>(partials, out);
}